// SparseVAE_34376918237634
// MI455X (gfx1250) — compile-verified
//
#include <hip/hip_runtime.h>
#include <math.h>

// ---------------------------------------------------------------------------
// SparseVAE forward for MI455X (gfx1250, wave32).
// 3x3 SAME convs -> implicit GEMM on V_WMMA_F32_16X16X4_F32 (exact fp32),
// as 9 (kh,kw) shift passes with K=Cin. All loads are UNCONDITIONAL
// (clamped addresses + value cndmask) so no saveexec/xcnt churn.
// One wave = 16 pixels (M) x Cout (N, up to 2 x 16-channel tiles).
// ---------------------------------------------------------------------------

typedef __attribute__((ext_vector_type(2))) float v2f;
typedef __attribute__((ext_vector_type(8))) float v8f;

#define BB   256
#define LATD 4
#define CC   32
#define HC   2
#define WC   16
#define NEGS 0.01f

__device__ __forceinline__ float lrelu_f(float v) {
    return v >= 0.0f ? v : NEGS * v;
}

template<int Cin, int Cout, int H, int W, bool LRELU>
__global__ __launch_bounds__(256) void conv3x3_wmma(
    const float* __restrict__ x, const float* __restrict__ wgt,
    const float* __restrict__ bias, float* __restrict__ y)
{
    constexpr int NT = (Cout + 15) / 16;   // N-tiles held per wave (1 or 2)
    constexpr int K9 = Cin * 9;            // OIHW flat stride per out-channel
    constexpr int WT = W / 16;
    constexpr int HW = H * W;
    constexpr int CinFull = Cin & ~3;      // K handled in full chunks of 4

    const int lane = threadIdx.x & 31;
    // Scalarize the wave id: tile indices live in SGPRs, bounds checks become
    // s_cbranch (EXEC untouched -> structurally all-ones at every WMMA).
    const int wv = __builtin_amdgcn_readfirstlane(
        blockIdx.x * 8 + (threadIdx.x >> 5));
    const int total = BB * H * WT;
    if (wv >= total) return;

    int t = wv;
    const int b  = t / (H * WT);  t -= b * (H * WT);
    const int h  = t / WT;
    const int wt = t - h * WT;
    const int w0 = wt << 4;

    const int m     = lane & 15;                 // pixel in tile   (A: M index)
    const int chalf = (lane >> 4) << 1;          // K sub-offset: 0 | 2
    const int co    = lane & 15;                 // channel in tile (B: N index)
    const int coc   = (Cout < 16) ? (co < Cout ? co : Cout - 1) : co;  // addr clamp

    v8f acc[NT];
    #pragma unroll
    for (int nt = 0; nt < NT; ++nt)
        acc[nt] = (v8f){0.f, 0.f, 0.f, 0.f, 0.f, 0.f, 0.f, 0.f};

    const float* xb = x + (size_t)b * Cin * HW + (size_t)chalf * HW;
    const float* wb = wgt + (size_t)coc * K9 + chalf * 9;

    #pragma unroll
    for (int kh = 0; kh < 3; ++kh) {
        const int hh = h + kh - 1;
        if (hh < 0 || hh >= H) continue;         // scalar branch (h in SGPR)
        #pragma unroll
        for (int kw = 0; kw < 3; ++kw) {
            const int  ww  = w0 + m + kw - 1;
            const bool inb = (unsigned)ww < (unsigned)W;
            const int  wwc = inb ? ww : 0;       // clamp address; mask value
            const float* xp = xb + (size_t)hh * W + wwc;
            const float* wp = wb + (kh * 3 + kw);

            // Full K chunks: unconditional loads, only the `inb` value mask.
            #pragma unroll
            for (int ci0 = 0; ci0 < CinFull; ci0 += 4) {
                v2f a;
                const float r0 = xp[(ci0    ) * HW];
                const float r1 = xp[(ci0 + 1) * HW];
                a.x = inb ? r0 : 0.f;
                a.y = inb ? r1 : 0.f;
                #pragma unroll
                for (int nt = 0; nt < NT; ++nt) {
                    v2f bf;
                    bf.x = wp[nt * 16 * K9 + (ci0    ) * 9];
                    bf.y = wp[nt * 16 * K9 + (ci0 + 1) * 9];
                    acc[nt] = __builtin_amdgcn_wmma_f32_16x16x4_f32(
                        false, a, false, bf, (short)0, acc[nt], false, false);
                }
            }

            // Tail chunk (only exists for Cin % 4 != 0, i.e. Cin == 6):
            // upper half-wave (chalf==2) would read K >= Cin -> clamp both
            // pointers back 2 channels and zero-mask A there.
            if constexpr ((Cin & 3) != 0) {
                const bool ok = (chalf == 0);
                const float* xq = ok ? xp : xp - 2 * HW;
                const float* wq = ok ? wp : wp - 2 * 9;
                v2f a;
                const float r0 = xq[(CinFull    ) * HW];
                const float r1 = xq[(CinFull + 1) * HW];
                a.x = (ok && inb) ? r0 : 0.f;
                a.y = (ok && inb) ? r1 : 0.f;
                #pragma unroll
                for (int nt = 0; nt < NT; ++nt) {
                    v2f bf;
                    bf.x = wq[nt * 16 * K9 + (CinFull    ) * 9];
                    bf.y = wq[nt * 16 * K9 + (CinFull + 1) * 9];
                    acc[nt] = __builtin_amdgcn_wmma_f32_16x16x4_f32(
                        false, a, false, bf, (short)0, acc[nt], false, false);
                }
            }
        }
    }

    // Epilogue: C/D layout -> lane covers M = mbase..mbase+7, contiguous in W.
    const int mbase = (lane >> 4) << 3;
    #pragma unroll
    for (int nt = 0; nt < NT; ++nt) {
        const int con = nt * 16 + co;
        if (con < Cout) {
            const float bv = bias[con];
            float r[8];
            #pragma unroll
            for (int i = 0; i < 8; ++i) {
                float v = acc[nt][i] + bv;
                r[i] = LRELU ? lrelu_f(v) : v;
            }
            float* yp = y + (((size_t)b * Cout + con) * H + h) * W + w0 + mbase;
            reinterpret_cast<float4*>(yp)[0] = make_float4(r[0], r[1], r[2], r[3]);
            reinterpret_cast<float4*>(yp)[1] = make_float4(r[4], r[5], r[6], r[7]);
        }
    }
}

// ---------------------------------------------------------------------------
// Non-overlapping max pool (kernel == stride), VALID.
// ---------------------------------------------------------------------------
template<int KH, int KW>
__global__ __launch_bounds__(256) void maxpool_k(
    const float* __restrict__ x, float* __restrict__ y,
    int Bn, int C, int H, int W)
{
    const int Ho = H / KH, Wo = W / KW;
    const size_t total = (size_t)Bn * C * Ho * Wo;
    size_t i = (size_t)blockIdx.x * blockDim.x + threadIdx.x;
    if (i >= total) return;
    size_t t = i;
    const int wo = t % Wo; t /= Wo;
    const int ho = t % Ho; t /= Ho;
    const int c  = t % C;
    const int b  = t / C;
    const float* p = x + (((size_t)b * C + c) * H + ho * KH) * W + (size_t)wo * KW;
    float mx = -INFINITY;
    #pragma unroll
    for (int r = 0; r < KH; ++r)
        #pragma unroll
        for (int s = 0; s < KW; ++s)
            mx = fmaxf(mx, p[(size_t)r * W + s]);
    y[i] = mx;
}

// ---------------------------------------------------------------------------
// Latent: per spatial cell n -> mu, logvar, z, kld_cell, decoder input row.
// ---------------------------------------------------------------------------
__global__ __launch_bounds__(256) void latent_k(
    const float* __restrict__ p3, const float* __restrict__ eps,
    const float* __restrict__ wmu, const float* __restrict__ bmu,
    const float* __restrict__ wlv, const float* __restrict__ blv,
    const float* __restrict__ wlin, const float* __restrict__ blin,
    float* __restrict__ mu_out, float* __restrict__ lv_out,
    float* __restrict__ z_out, float* __restrict__ kldcell,
    float* __restrict__ dec)
{
    const int n = blockIdx.x * blockDim.x + threadIdx.x;
    const int N = BB * HC * WC;
    if (n >= N) return;
    const int cell = n % (HC * WC);
    const int b    = n / (HC * WC);
    const int h    = cell / WC;
    const int w    = cell % WC;

    float f[CC];
    #pragma unroll
    for (int c = 0; c < CC; ++c)
        f[c] = p3[(((size_t)b * CC + c) * HC + h) * WC + w];

    float z[LATD];
    float kl = 0.0f;
    #pragma unroll
    for (int j = 0; j < LATD; ++j) {
        float mm = bmu[j], l = blv[j];
        #pragma unroll
        for (int c = 0; c < CC; ++c) {
            mm += f[c] * wmu[c * LATD + j];
            l  += f[c] * wlv[c * LATD + j];
        }
        const float zz = mm + eps[(size_t)n * LATD + j] * expf(0.5f * l);
        mu_out[(size_t)n * LATD + j] = mm;
        lv_out[(size_t)n * LATD + j] = l;
        z_out [(size_t)n * LATD + j] = zz;
        z[j] = zz;
        kl += 1.0f + l - mm * mm - expf(l);
    }
    kldcell[n] = -0.5f * kl;

    #pragma unroll
    for (int co = 0; co < CC; ++co) {
        float d = blin[co];
        #pragma unroll
        for (int j = 0; j < LATD; ++j) d += z[j] * wlin[j * CC + co];
        dec[(((size_t)b * CC + co) * HC + h) * WC + w] = d;
    }
}

// kld[b] = mean over the 32 cells of batch b
__global__ __launch_bounds__(256) void kld_k(
    const float* __restrict__ kldcell, float* __restrict__ kld)
{
    const int b = blockIdx.x * blockDim.x + threadIdx.x;
    if (b >= BB) return;
    float s = 0.0f;
    #pragma unroll
    for (int i = 0; i < HC * WC; ++i) s += kldcell[b * (HC * WC) + i];
    kld[b] = s / (float)(HC * WC);
}

// ---------------------------------------------------------------------------
// Transposed conv, stride == kernel (non-overlapping), fused leaky ReLU.
// w: (Cin, Cout, KH, KW). out[b,co,h*KH+r,w*KW+s] = b[co] + sum_c x*w
// ---------------------------------------------------------------------------
template<int Cin, int Cout, int KH, int KW>
__global__ __launch_bounds__(256) void invconv_k(
    const float* __restrict__ x, const float* __restrict__ w,
    const float* __restrict__ bias, float* __restrict__ y, int Bn, int H, int W)
{
    const int Ho = H * KH, Wo = W * KW;
    const size_t total = (size_t)Bn * Cout * Ho * Wo;
    size_t i = (size_t)blockIdx.x * blockDim.x + threadIdx.x;
    if (i >= total) return;
    size_t t = i;
    const int wo = t % Wo; t /= Wo;
    const int ho = t % Ho; t /= Ho;
    const int co = t % Cout;
    const int b  = t / Cout;
    const int h = ho / KH, r = ho % KH;
    const int wc = wo / KW, s = wo % KW;

    float acc = bias[co];
    #pragma unroll
    for (int c = 0; c < Cin; ++c)
        acc += x[(((size_t)b * Cin + c) * H + h) * W + wc] *
               w[(((size_t)c * Cout + co) * KH + r) * KW + s];
    y[i] = lrelu_f(acc);
}

// ---------------------------------------------------------------------------

extern "C" void kernel_launch(void* const* d_in, const int* in_sizes, int n_in,
                              void* d_out, int out_size, void* d_ws, size_t ws_size,
                              hipStream_t stream)
{
    const float* x    = (const float*)d_in[0];
    const float* eps  = (const float*)d_in[1];
    const float* w1   = (const float*)d_in[2];  const float* b1  = (const float*)d_in[3];
    const float* w2   = (const float*)d_in[4];  const float* b2  = (const float*)d_in[5];
    const float* w3   = (const float*)d_in[6];  const float* b3  = (const float*)d_in[7];
    const float* wmu  = (const float*)d_in[8];  const float* bmu = (const float*)d_in[9];
    const float* wlv  = (const float*)d_in[10]; const float* blv = (const float*)d_in[11];
    const float* wlin = (const float*)d_in[12]; const float* blin= (const float*)d_in[13];
    const float* wi1  = (const float*)d_in[14]; const float* bi1 = (const float*)d_in[15];
    const float* w4   = (const float*)d_in[16]; const float* b4  = (const float*)d_in[17];
    const float* wi2  = (const float*)d_in[18]; const float* bi2 = (const float*)d_in[19];
    const float* w5   = (const float*)d_in[20]; const float* b5  = (const float*)d_in[21];
    const float* wi3  = (const float*)d_in[22]; const float* bi3 = (const float*)d_in[23];
    const float* w6   = (const float*)d_in[24]; const float* b6  = (const float*)d_in[25];

    // Output slices (flat, in reference return order)
    float* out = (float*)d_out;                       // (B,6,100,128) = 19,660,800
    float* mu  = out + (size_t)19660800;              // (8192,4)
    float* lv  = mu  + (size_t)32768;                 // (8192,4)
    float* kld = lv  + (size_t)32768;                 // (256,)
    float* z   = kld + (size_t)256;                   // (8192,4)

    // Workspace carve-out (floats), ping-pong reuse
    float* ws  = (float*)d_ws;
    float* A   = ws;                                  // 26,214,400 : big ping buffer
    float* Bb  = A   + (size_t)26214400;              //  6,553,600 : p1 / h4 / h5
    float* P2  = Bb  + (size_t)6553600;               //  1,310,720
    float* P3  = P2  + (size_t)1310720;               //    262,144
    float* DEC = P3  + (size_t)262144;                //    262,144
    float* KC  = DEC + (size_t)262144;                //      8,192

    auto ew_blocks = [](size_t total) { return (int)((total + 255) / 256); };
    auto cv_blocks = [](int H, int W)  { return (BB * H * (W / 16) + 7) / 8; };

    // ---- encoder ----
    conv3x3_wmma<6, 8, 100, 128, true><<<cv_blocks(100, 128), 256, 0, stream>>>(x, w1, b1, A);
    maxpool_k<2, 2><<<ew_blocks((size_t)BB * 8 * 50 * 64), 256, 0, stream>>>(A, Bb, BB, 8, 100, 128);
    conv3x3_wmma<8, 16, 50, 64, true><<<cv_blocks(50, 64), 256, 0, stream>>>(Bb, w2, b2, A);
    maxpool_k<5, 2><<<ew_blocks((size_t)BB * 16 * 10 * 32), 256, 0, stream>>>(A, P2, BB, 16, 50, 64);
    conv3x3_wmma<16, 32, 10, 32, true><<<cv_blocks(10, 32), 256, 0, stream>>>(P2, w3, b3, A);
    maxpool_k<5, 2><<<ew_blocks((size_t)BB * 32 * 2 * 16), 256, 0, stream>>>(A, P3, BB, 32, 10, 32);

    // ---- latent + KLD ----
    latent_k<<<ew_blocks((size_t)BB * HC * WC), 256, 0, stream>>>(
        P3, eps, wmu, bmu, wlv, blv, wlin, blin, mu, lv, z, KC, DEC);
    kld_k<<<1, 256, 0, stream>>>(KC, kld);

    // ---- decoder ----
    invconv_k<32, 32, 5, 2><<<ew_blocks((size_t)BB * 32 * 10 * 32), 256, 0, stream>>>(
        DEC, wi1, bi1, A, BB, 2, 16);                                   // u1
    conv3x3_wmma<32, 16, 10, 32, true><<<cv_blocks(10, 32), 256, 0, stream>>>(A, w4, b4, Bb);
    invconv_k<16, 16, 5, 2><<<ew_blocks((size_t)BB * 16 * 50 * 64), 256, 0, stream>>>(
        Bb, wi2, bi2, A, BB, 10, 32);                                   // u2
    conv3x3_wmma<16, 8, 50, 64, true><<<cv_blocks(50, 64), 256, 0, stream>>>(A, w5, b5, Bb);
    invconv_k<8, 8, 2, 2><<<ew_blocks((size_t)BB * 8 * 100 * 128), 256, 0, stream>>>(
        Bb, wi3, bi3, A, BB, 50, 64);                                   // u3
    conv3x3_wmma<8, 6, 100, 128, false><<<cv_blocks(100, 128), 256, 0, stream>>>(A, w6, b6, out);
}